// recat_55860344651791
// MI455X (gfx1250) — compile-verified
//
#include <hip/hip_runtime.h>
#include <math.h>

typedef __attribute__((ext_vector_type(2))) float v2f;
typedef __attribute__((ext_vector_type(4))) float v4f;
typedef __attribute__((ext_vector_type(8))) float v8f;

#define D_DIM 1024
#define B_DIM 2048
#define M_MOL 8
#define NPAIR 28

// np.triu_indices(8, 1)
__device__ __constant__ unsigned char c_I[NPAIR] =
    {0,0,0,0,0,0,0, 1,1,1,1,1,1, 2,2,2,2,2, 3,3,3,3, 4,4,4, 5,5, 6};
__device__ __constant__ unsigned char c_J[NPAIR] =
    {1,2,3,4,5,6,7, 2,3,4,5,6,7, 3,4,5,6,7, 4,5,6,7, 5,6,7, 6,7, 7};

static __device__ inline v8f wmma4(v2f a, v2f b, v8f c) {
  // D = A(16x4 f32) * B(4x16 f32) + C(16x16 f32)
  return __builtin_amdgcn_wmma_f32_16x16x4_f32(false, a, false, b, (short)0, c,
                                               false, false);
}

// Packed-G layout: row e of G^T is stored as two half-streams so that a lane
// (which owns K-values d%4 in {2h, 2h+1}) reads a fully contiguous stream:
//   Gpk[e*1024 + h*512 + (d>>2)*2 + (d&1)]  where h = (d>>1)&1
// One global_load_b128 then feeds TWO K-steps of the WMMA B operand.

// ---------------------------------------------------------------------------
// Kernel A: Gpk = packed( (Wq @ Wk^T)^T ) i.e. Gpk row e holds sum_x Wk[e,x]Wq[d,x]
// One 16x16 output tile per wave; 4096 waves total. (2.1 GFLOP, negligible)
// ---------------------------------------------------------------------------
__global__ __launch_bounds__(256) void gemm_gt(const float* __restrict__ Wq,
                                               const float* __restrict__ Wk,
                                               float* __restrict__ Gpk) {
  const int wave = (int)((blockIdx.x * blockDim.x + threadIdx.x) >> 5);
  const int lane = threadIdx.x & 31;
  const int lm = lane & 15;
  const int half = lane >> 4;
  const int tm = wave >> 6;   // row tile (e)
  const int tn = wave & 63;   // col tile (d)

  const float* arow = Wk + (size_t)(tm * 16 + lm) * D_DIM + 2 * half;
  const float* brow = Wq + (size_t)(tn * 16 + lm) * D_DIM + 2 * half;

  v8f c = {};
#pragma unroll 4
  for (int x0 = 0; x0 < D_DIM; x0 += 4) {
    v2f a = *(const v2f*)(arow + x0);
    v2f b = *(const v2f*)(brow + x0);
    c = wmma4(a, b, c);
  }
  // C layout: VGPR v, lane (half,lm) -> row = v + 8*half, col = lm
  // d = tn*16 + lm  ->  packed in-row offset:
  const int colOff = ((lm >> 1) & 1) * 512 + (tn * 4 + (lm >> 2)) * 2 + (lm & 1);
#pragma unroll
  for (int v = 0; v < 8; ++v)
    Gpk[(size_t)(tm * 16 + v + 8 * half) * D_DIM + colOff] = c[v];
}

// ---------------------------------------------------------------------------
// Kernel B: one workgroup (256 threads = 8 wave32) per batch b.
//   F = [p_b ; r_b]  (16 x 1024, rows read straight from global)
//   S = F * G * F^T  (16x16): W = F*G column-tiles (4 concurrent WMMA chains,
//   packed-G b128 loads), each tile immediately contracted against F^T.
//   Then: 36x36 score expansion both directions, softmax, mean -> att outputs;
//   effective 8-row pooling weights -> reaction -> W_pred projection.
// ---------------------------------------------------------------------------
__global__ __launch_bounds__(256) void attn_batch(
    const float* __restrict__ rF, const float* __restrict__ pF,
    const float* __restrict__ rN, const float* __restrict__ pN,
    const float* __restrict__ Gpk, const float* __restrict__ Wp,
    const float* __restrict__ bp, float* __restrict__ out) {
  __shared__ float sW[8][256];    // per-wave W-tile staging (C->A relayout)
  __shared__ float sS[8][256];    // per-wave partial S for reduction
  __shared__ float sRows[72][36]; // softmax rows
  __shared__ float sAtt[2][36];
  __shared__ float sWeff[2][8];
  __shared__ float sRed[256 * 4]; // prediction reduction

  const int b = blockIdx.x;
  const int tid = threadIdx.x;
  const int wave = tid >> 5;
  const int lane = tid & 31;
  const int lm = lane & 15;
  const int half = lane >> 4;

  // F rows 0..7 = p_b, rows 8..15 = r_b; lane lm owns row lm
  const float* Frow =
      (lm < M_MOL) ? (pF + ((size_t)lm * B_DIM + b) * D_DIM)
                   : (rF + ((size_t)(lm - M_MOL) * B_DIM + b) * D_DIM);
  const float* fa = Frow + 2 * half;

  v8f sacc = {};
#pragma unroll 1
  for (int i = 0; i < 2; ++i) {
    // this wave's e-tiles: wave + 8*(4i+j), j = 0..3 (4 concurrent chains)
    int e[4];
    const float* g[4];
#pragma unroll
    for (int j = 0; j < 4; ++j) {
      e[j] = (wave + 8 * (4 * i + j)) * 16;
      g[j] = Gpk + (size_t)(e[j] + lm) * D_DIM + half * 512;
    }
    v8f c0 = {}, c1 = {}, c2 = {}, c3 = {};
#pragma unroll 1
    for (int d0 = 0; d0 < D_DIM; d0 += 8) {
      // A operand for K-steps d0 and d0+4 (shared by all 4 chains)
      v2f a0 = *(const v2f*)(fa + d0);
      v2f a1 = *(const v2f*)(fa + d0 + 4);
      // packed-G: one b128 per chain covers both K-steps
      v4f g0 = *(const v4f*)(g[0] + (d0 >> 1));
      v4f g1 = *(const v4f*)(g[1] + (d0 >> 1));
      v4f g2 = *(const v4f*)(g[2] + (d0 >> 1));
      v4f g3 = *(const v4f*)(g[3] + (d0 >> 1));
      c0 = wmma4(a0, __builtin_shufflevector(g0, g0, 0, 1), c0);
      c1 = wmma4(a0, __builtin_shufflevector(g1, g1, 0, 1), c1);
      c2 = wmma4(a0, __builtin_shufflevector(g2, g2, 0, 1), c2);
      c3 = wmma4(a0, __builtin_shufflevector(g3, g3, 0, 1), c3);
      c0 = wmma4(a1, __builtin_shufflevector(g0, g0, 2, 3), c0);
      c1 = wmma4(a1, __builtin_shufflevector(g1, g1, 2, 3), c1);
      c2 = wmma4(a1, __builtin_shufflevector(g2, g2, 2, 3), c2);
      c3 = wmma4(a1, __builtin_shufflevector(g3, g3, 2, 3), c3);
    }
    // --- contract each W tile into S: S += W_tile * F[:, e:e+16]^T ---
#pragma unroll
    for (int j = 0; j < 4; ++j) {
      const v8f cj = (j == 0) ? c0 : (j == 1) ? c1 : (j == 2) ? c2 : c3;
#pragma unroll
      for (int v = 0; v < 8; ++v) sW[wave][(v + 8 * half) * 16 + lm] = cj[v];
#pragma unroll
      for (int kk = 0; kk < 4; ++kk) {
        v2f a2 = *(const v2f*)(&sW[wave][lm * 16 + kk * 4 + 2 * half]);
        v2f b2 = *(const v2f*)(Frow + e[j] + kk * 4 + 2 * half);
        sacc = wmma4(a2, b2, sacc);
      }
    }
  }
#pragma unroll
  for (int v = 0; v < 8; ++v) sS[wave][(v + 8 * half) * 16 + lm] = sacc[v];
  __syncthreads();

  {  // cross-wave reduce: S_full in sS[0][i*16+j]
    float s = 0.f;
#pragma unroll
    for (int w = 0; w < 8; ++w) s += sS[w][tid];
    sS[0][tid] = s;
  }
  __syncthreads();

  // ---- expand to 36x36 scores (both directions), softmax per row ----
  const float scale = 0.03125f;  // 1/sqrt(1024)
  if (tid < 72) {
    const int dir = (tid >= 36) ? 1 : 0;  // 0: att_reactant (q=p,k=r)
    const int q = dir ? tid - 36 : tid;
    int qa0 = q, qa1 = q, qn = 1;
    if (q >= 8) { qa0 = c_I[q - 8]; qa1 = c_J[q - 8]; qn = 2; }
    float row[36];
    float mx = -3.4e38f;
#pragma unroll
    for (int k = 0; k < 36; ++k) {
      int ka0 = k, ka1 = k, kn = 1;
      if (k >= 8) { ka0 = c_I[k - 8]; ka1 = c_J[k - 8]; kn = 2; }
      float s = 0.f;
      for (int u = 0; u < qn; ++u) {
        const int a = u ? qa1 : qa0;
        for (int v = 0; v < kn; ++v) {
          const int cc = v ? ka1 : ka0;
          s += (dir == 0) ? sS[0][a * 16 + 8 + cc] : sS[0][(8 + a) * 16 + cc];
        }
      }
      s *= scale;
      row[k] = s;
      mx = fmaxf(mx, s);
    }
    float sum = 0.f;
#pragma unroll
    for (int k = 0; k < 36; ++k) { float e = __expf(row[k] - mx); row[k] = e; sum += e; }
    const float inv = 1.f / sum;
#pragma unroll
    for (int k = 0; k < 36; ++k) sRows[tid][k] = row[k] * inv;
  }
  __syncthreads();

  if (tid < 72) {  // mean over queries -> att[36]; write outputs
    const int dir = (tid >= 36) ? 1 : 0;
    const int k = dir ? tid - 36 : tid;
    float a = 0.f;
#pragma unroll
    for (int q = 0; q < 36; ++q) a += sRows[dir * 36 + q][k];
    a *= (1.f / 36.f);
    sAtt[dir][k] = a;
    const size_t base = dir ? (size_t)(B_DIM * 4 + B_DIM * 36) : (size_t)(B_DIM * 4);
    out[base + (size_t)b * 36 + k] = a;
  }
  __syncthreads();

  if (tid < 16) {  // effective per-molecule pooling weights
    const int dir = tid >> 3, m = tid & 7;
    float w = sAtt[dir][m];
#pragma unroll
    for (int pr = 0; pr < NPAIR; ++pr)
      if (c_I[pr] == m || c_J[pr] == m) w += sAtt[dir][8 + pr];
    sWeff[dir][m] = w;  // dir0 -> reactant (r rows), dir1 -> product (p rows)
  }
  __syncthreads();

  // ---- pooling + prediction: out[b] = reaction @ W_pred^T + b_pred ----
  {
    const int d = tid * 4;
    float reac[4] = {0.f, 0.f, 0.f, 0.f};
#pragma unroll
    for (int m = 0; m < M_MOL; ++m) {
      const size_t off = ((size_t)m * B_DIM + b) * D_DIM + d;
      v4f rv = *(const v4f*)(rF + off);
      v4f rnv = *(const v4f*)(rN + off);
      v4f pv = *(const v4f*)(pF + off);
      v4f pnv = *(const v4f*)(pN + off);
      const float wr = sWeff[0][m], wp = sWeff[1][m];
#pragma unroll
      for (int x = 0; x < 4; ++x)
        reac[x] += wr * (rv[x] + rnv[x]) - wp * (pv[x] + pnv[x]);
    }
#pragma unroll
    for (int o = 0; o < 4; ++o) {
      float s = 0.f;
#pragma unroll
      for (int x = 0; x < 4; ++x) s += reac[x] * Wp[(size_t)o * D_DIM + d + x];
      sRed[tid * 4 + o] = s;
    }
  }
  __syncthreads();
  if (tid < 4) {
    float s = bp[tid];
    for (int t = 0; t < 256; ++t) s += sRed[t * 4 + tid];
    out[(size_t)b * 4 + tid] = s;
  }
}

extern "C" void kernel_launch(void* const* d_in, const int* in_sizes, int n_in,
                              void* d_out, int out_size, void* d_ws,
                              size_t ws_size, hipStream_t stream) {
  (void)in_sizes; (void)n_in; (void)out_size; (void)ws_size;
  const float* rF = (const float*)d_in[0];
  const float* pF = (const float*)d_in[1];
  const float* rN = (const float*)d_in[2];
  const float* pN = (const float*)d_in[3];
  const float* Wq = (const float*)d_in[4];
  const float* Wk = (const float*)d_in[5];
  const float* Wp = (const float*)d_in[6];
  const float* bp = (const float*)d_in[7];
  float* out = (float*)d_out;
  float* Gpk = (float*)d_ws;  // 1024*1024 f32 = 4 MB workspace (packed G^T)

  gemm_gt<<<512, 256, 0, stream>>>(Wq, Wk, Gpk);
  attn_batch<<<B_DIM, 256, 0, stream>>>(rF, pF, rN, pN, Gpk, Wp, bp, out);
}